// RWKV_26757646254612
// MI455X (gfx1250) — compile-verified
//
#include <hip/hip_runtime.h>
#include <math.h>

typedef __attribute__((ext_vector_type(2))) float v2f;
typedef __attribute__((ext_vector_type(8))) float v8f;
typedef __attribute__((ext_vector_type(4))) int   v4i;

#if __has_builtin(__builtin_amdgcn_global_load_async_to_lds_b128) && \
    __has_builtin(__builtin_amdgcn_s_wait_asynccnt)
#define USE_ASYNC_LDS 1
typedef __attribute__((address_space(1))) v4i gv4i;   // global 16B vector
typedef __attribute__((address_space(3))) v4i lv4i;   // LDS 16B vector
#else
#define USE_ASYNC_LDS 0
#endif

#define LAYERS 4
#define CDIM   1024
#define ADIM   1024
#define FFDIM  4096
#define VDIM   32000
#define BDIM   2
#define TDIM   1024
#define MROWS  (BDIM * TDIM)   // 2048

// ---------------------------------------------------------------------------
// Embedding gather + LayerNorm(ln0). One block per row, 256 threads, C=1024.
// ---------------------------------------------------------------------------
__global__ __launch_bounds__(256)
void embed_ln_kernel(const int* __restrict__ tokens, const float* __restrict__ emb,
                     const float* __restrict__ w, const float* __restrict__ b,
                     float* __restrict__ out) {
    const int row = blockIdx.x;
    const int tok = tokens[row];
    const float* e = emb + (size_t)tok * CDIM;
    __shared__ float s1[256], s2[256];
    float vals[CDIM / 256];
    float lsum = 0.f, lsq = 0.f;
    #pragma unroll
    for (int i = 0; i < CDIM / 256; ++i) {
        float x = e[threadIdx.x + i * 256];
        vals[i] = x; lsum += x; lsq += x * x;
    }
    s1[threadIdx.x] = lsum; s2[threadIdx.x] = lsq;
    __syncthreads();
    for (int s = 128; s > 0; s >>= 1) {
        if (threadIdx.x < s) {
            s1[threadIdx.x] += s1[threadIdx.x + s];
            s2[threadIdx.x] += s2[threadIdx.x + s];
        }
        __syncthreads();
    }
    const float mean = s1[0] * (1.f / CDIM);
    const float var  = s2[0] * (1.f / CDIM) - mean * mean;
    const float inv  = rsqrtf(var + 1e-5f);
    #pragma unroll
    for (int i = 0; i < CDIM / 256; ++i) {
        int c = threadIdx.x + i * 256;
        out[(size_t)row * CDIM + c] = (vals[i] - mean) * inv * w[c] + b[c];
    }
}

// ---------------------------------------------------------------------------
// Row LayerNorm: out = LN(in) * w + b. One block per row.
// ---------------------------------------------------------------------------
__global__ __launch_bounds__(256)
void ln_kernel(const float* __restrict__ in, const float* __restrict__ w,
               const float* __restrict__ b, float* __restrict__ out) {
    const int row = blockIdx.x;
    const float* x = in + (size_t)row * CDIM;
    __shared__ float s1[256], s2[256];
    float vals[CDIM / 256];
    float lsum = 0.f, lsq = 0.f;
    #pragma unroll
    for (int i = 0; i < CDIM / 256; ++i) {
        float v = x[threadIdx.x + i * 256];
        vals[i] = v; lsum += v; lsq += v * v;
    }
    s1[threadIdx.x] = lsum; s2[threadIdx.x] = lsq;
    __syncthreads();
    for (int s = 128; s > 0; s >>= 1) {
        if (threadIdx.x < s) {
            s1[threadIdx.x] += s1[threadIdx.x + s];
            s2[threadIdx.x] += s2[threadIdx.x + s];
        }
        __syncthreads();
    }
    const float mean = s1[0] * (1.f / CDIM);
    const float var  = s2[0] * (1.f / CDIM) - mean * mean;
    const float inv  = rsqrtf(var + 1e-5f);
    #pragma unroll
    for (int i = 0; i < CDIM / 256; ++i) {
        int c = threadIdx.x + i * 256;
        out[(size_t)row * CDIM + c] = (vals[i] - mean) * inv * w[c] + b[c];
    }
}

// ---------------------------------------------------------------------------
// Token-shift mixing: mk = xn*tk + xx*(1-tk)  (xx = xn shifted by one token,
// zero at t==0). Optionally also produces the v-mix.
// ---------------------------------------------------------------------------
__global__ __launch_bounds__(256)
void mix_kernel(const float* __restrict__ xn,
                const float* __restrict__ tk, const float* __restrict__ tv,
                const float* __restrict__ tr,
                float* __restrict__ mk, float* __restrict__ mv,
                float* __restrict__ mr, int has_v) {
    const int idx = blockIdx.x * 256 + threadIdx.x;       // over MROWS*CDIM
    const int c   = idx & (CDIM - 1);
    const int row = idx >> 10;                            // CDIM == 1024
    const int t   = row & (TDIM - 1);                     // TDIM == 1024
    const float cur  = xn[idx];
    const float prev = (t == 0) ? 0.f : xn[idx - CDIM];
    const float k_ = tk[c];
    mk[idx] = cur * k_ + prev * (1.f - k_);
    if (has_v) {
        const float v_ = tv[c];
        mv[idx] = cur * v_ + prev * (1.f - v_);
    }
    const float r_ = tr[c];
    mr[idx] = cur * r_ + prev * (1.f - r_);
}

// ---------------------------------------------------------------------------
// WKV recurrence (numerically-stable exp state). One thread per (batch,chan);
// sequential over T. Fuses sry = sigmoid(r) * y.
// ---------------------------------------------------------------------------
__global__ __launch_bounds__(256)
void wkv_kernel(const float* __restrict__ td, const float* __restrict__ tf,
                const float* __restrict__ k, const float* __restrict__ v,
                const float* __restrict__ r, float* __restrict__ sry) {
    const int idx = blockIdx.x * 256 + threadIdx.x;       // 0 .. B*A-1
    const int a = idx & (ADIM - 1);
    const int b = idx >> 10;                              // ADIM == 1024
    const float w = -expf(td[a]);
    const float u = tf[a];
    float aa = 0.f, bb = 0.f, pp = -1e38f;
    const size_t base = (size_t)b * TDIM * ADIM + a;
    for (int t = 0; t < TDIM; ++t) {
        const size_t o = base + (size_t)t * ADIM;
        const float kt = k[o], vt = v[o];
        const float ww = u + kt;
        const float p  = fmaxf(pp, ww);
        const float e1 = expf(pp - p);
        const float e2 = expf(ww - p);
        const float y  = (e1 * aa + e2 * vt) / (e1 * bb + e2);
        const float ww2 = pp + w;
        const float p2  = fmaxf(ww2, kt);
        const float e1b = expf(ww2 - p2);
        const float e2b = expf(kt - p2);
        aa = e1b * aa + e2b * vt;
        bb = e1b * bb + e2b;
        pp = p2;
        const float rt = r[o];
        sry[o] = y * (1.f / (1.f + expf(-rt)));
    }
}

// ---------------------------------------------------------------------------
// x += g * kv (FFN residual update)
// ---------------------------------------------------------------------------
__global__ __launch_bounds__(256)
void fma_add_kernel(float* __restrict__ x, const float* __restrict__ g,
                    const float* __restrict__ kv) {
    const int idx = blockIdx.x * 256 + threadIdx.x;
    x[idx] += g[idx] * kv[idx];
}

// ---------------------------------------------------------------------------
// WMMA fp32 GEMM: Y[M,N] = epi( X[M,K] * W[N,K]^T ).
// Block = 256 threads (8 waves), tile 64(M) x 128(N), K blocked by 32.
// Staging:
//  * async path  : GLOBAL_LOAD_ASYNC_TO_LDS_B128 double-buffered ping-pong,
//                  ASYNCcnt-ordered (wait <=6 drains the current chunk while
//                  the next chunk's DMA stays in flight under the WMMAs).
//  * fallback    : global_load_b128 -> regs -> ds_store_b128 with register
//                  prefetch of the next chunk.
// LDS rows padded to 36 floats so the (row=r, col=k+2*half) fragment reads
// map every lane to a distinct bank pair (conflict-free ds_load_b64).
// Epilogue: 0 = store, 1 = sigmoid, 2 = relu^2, 3 = Y += acc.
// Requires M%64==0, N%128==0, K%32==0 (all shapes here comply) -> no
// divergence: EXEC stays all-ones as WMMA requires.
// ---------------------------------------------------------------------------
__global__ __launch_bounds__(256)
void gemm_wmma_kernel(const float* __restrict__ X, const float* __restrict__ W,
                      float* __restrict__ Y, int M, int N, int K, int epi) {
    constexpr int KB   = 32;   // K chunk
    constexpr int LSTR = 36;   // padded LDS row stride (floats)

    const int tid  = threadIdx.x;
    const int lane = tid & 31;
    const int wave = tid >> 5;
    const int half = lane >> 4;        // 0: lanes 0-15, 1: lanes 16-31
    const int r    = lane & 15;
    const int n0   = blockIdx.x * 128;
    const int m0   = blockIdx.y * 64;
    const int koff = 2 * half;

    // Cooperative staging map: KB=32 floats/row = 8 float4 per row.
    // W tile: 128 rows * 8 = 1024 float4 -> 4 per thread.
    // X tile:  64 rows * 8 =  512 float4 -> 2 per thread.
    int wrow[4], wcol[4], xrow[2], xcol[2];
    #pragma unroll
    for (int i = 0; i < 4; ++i) { int f = tid + 256 * i; wrow[i] = f >> 3; wcol[i] = (f & 7) * 4; }
    #pragma unroll
    for (int i = 0; i < 2; ++i) { int f = tid + 256 * i; xrow[i] = f >> 3; xcol[i] = (f & 7) * 4; }

    v8f acc0 = {0.f,0.f,0.f,0.f,0.f,0.f,0.f,0.f};
    v8f acc1 = acc0, acc2 = acc0, acc3 = acc0;

#if USE_ASYNC_LDS
    __shared__ float xls[2][64  * LSTR];
    __shared__ float wls[2][128 * LSTR];

    // Issue chunk 0 into buffer 0 (6 async DMAs per thread, ordered).
    #pragma unroll
    for (int i = 0; i < 4; ++i)
        __builtin_amdgcn_global_load_async_to_lds_b128(
            (gv4i*)(W + (size_t)(n0 + wrow[i]) * K + wcol[i]),
            (lv4i*)&wls[0][wrow[i] * LSTR + wcol[i]], 0, 0);
    #pragma unroll
    for (int i = 0; i < 2; ++i)
        __builtin_amdgcn_global_load_async_to_lds_b128(
            (gv4i*)(X + (size_t)(m0 + xrow[i]) * K + xcol[i]),
            (lv4i*)&xls[0][xrow[i] * LSTR + xcol[i]], 0, 0);

    int buf = 0;
    for (int kb = 0; kb < K; kb += KB) {
        if (kb + KB < K) {                 // DMA next chunk into other buffer
            const int kn = kb + KB;
            const int nb = buf ^ 1;
            #pragma unroll
            for (int i = 0; i < 4; ++i)
                __builtin_amdgcn_global_load_async_to_lds_b128(
                    (gv4i*)(W + (size_t)(n0 + wrow[i]) * K + kn + wcol[i]),
                    (lv4i*)&wls[nb][wrow[i] * LSTR + wcol[i]], 0, 0);
            #pragma unroll
            for (int i = 0; i < 2; ++i)
                __builtin_amdgcn_global_load_async_to_lds_b128(
                    (gv4i*)(X + (size_t)(m0 + xrow[i]) * K + kn + xcol[i]),
                    (lv4i*)&xls[nb][xrow[i] * LSTR + xcol[i]], 0, 0);
            __builtin_amdgcn_s_wait_asynccnt(6);   // current chunk drained
        } else {
            __builtin_amdgcn_s_wait_asynccnt(0);
        }
        __syncthreads();                   // every wave's DMA data visible

        const float* wp  = &wls[buf][(wave * 16 + r) * LSTR + koff];
        const float* xp0 = &xls[buf][r * LSTR + koff];
        #pragma unroll
        for (int ks = 0; ks < KB; ks += 4) {
            const v2f b  = *(const v2f*)(wp  + ks);
            const v2f a0 = *(const v2f*)(xp0 + ks);
            const v2f a1 = *(const v2f*)(xp0 + 16 * LSTR + ks);
            const v2f a2 = *(const v2f*)(xp0 + 32 * LSTR + ks);
            const v2f a3 = *(const v2f*)(xp0 + 48 * LSTR + ks);
            acc0 = __builtin_amdgcn_wmma_f32_16x16x4_f32(false, a0, false, b, (short)0, acc0, false, false);
            acc1 = __builtin_amdgcn_wmma_f32_16x16x4_f32(false, a1, false, b, (short)0, acc1, false, false);
            acc2 = __builtin_amdgcn_wmma_f32_16x16x4_f32(false, a2, false, b, (short)0, acc2, false, false);
            acc3 = __builtin_amdgcn_wmma_f32_16x16x4_f32(false, a3, false, b, (short)0, acc3, false, false);
        }
        __syncthreads();                   // all readers done before reusing buf
        buf ^= 1;
    }
#else
    __shared__ float xls[64  * LSTR];
    __shared__ float wls[128 * LSTR];

    float4 wreg[4], xreg[2];
    #pragma unroll
    for (int i = 0; i < 4; ++i)
        wreg[i] = *(const float4*)(W + (size_t)(n0 + wrow[i]) * K + wcol[i]);
    #pragma unroll
    for (int i = 0; i < 2; ++i)
        xreg[i] = *(const float4*)(X + (size_t)(m0 + xrow[i]) * K + xcol[i]);

    const float* wp  = wls + (wave * 16 + r) * LSTR + koff;
    const float* xp0 = xls + r * LSTR + koff;

    for (int kb = 0; kb < K; kb += KB) {
        __syncthreads();                       // prior chunk's readers done
        #pragma unroll
        for (int i = 0; i < 4; ++i)
            *(float4*)(wls + wrow[i] * LSTR + wcol[i]) = wreg[i];
        #pragma unroll
        for (int i = 0; i < 2; ++i)
            *(float4*)(xls + xrow[i] * LSTR + xcol[i]) = xreg[i];
        __syncthreads();                       // tile visible to all waves

        if (kb + KB < K) {                     // prefetch next chunk (uniform)
            const int kn = kb + KB;
            #pragma unroll
            for (int i = 0; i < 4; ++i)
                wreg[i] = *(const float4*)(W + (size_t)(n0 + wrow[i]) * K + kn + wcol[i]);
            #pragma unroll
            for (int i = 0; i < 2; ++i)
                xreg[i] = *(const float4*)(X + (size_t)(m0 + xrow[i]) * K + kn + xcol[i]);
        }

        #pragma unroll
        for (int ks = 0; ks < KB; ks += 4) {
            const v2f b  = *(const v2f*)(wp  + ks);
            const v2f a0 = *(const v2f*)(xp0 + ks);
            const v2f a1 = *(const v2f*)(xp0 + 16 * LSTR + ks);
            const v2f a2 = *(const v2f*)(xp0 + 32 * LSTR + ks);
            const v2f a3 = *(const v2f*)(xp0 + 48 * LSTR + ks);
            acc0 = __builtin_amdgcn_wmma_f32_16x16x4_f32(false, a0, false, b, (short)0, acc0, false, false);
            acc1 = __builtin_amdgcn_wmma_f32_16x16x4_f32(false, a1, false, b, (short)0, acc1, false, false);
            acc2 = __builtin_amdgcn_wmma_f32_16x16x4_f32(false, a2, false, b, (short)0, acc2, false, false);
            acc3 = __builtin_amdgcn_wmma_f32_16x16x4_f32(false, a3, false, b, (short)0, acc3, false, false);
        }
    }
#endif

    const int col = n0 + wave * 16 + r;
    #define STORE_TILE(ACC, MT)                                              \
    {                                                                        \
        _Pragma("unroll")                                                    \
        for (int j = 0; j < 8; ++j) {                                        \
            const int row = m0 + (MT) * 16 + j + 8 * half;                   \
            float val = (ACC)[j];                                            \
            float* yp = Y + (size_t)row * N + col;                           \
            if (epi == 1)      *yp = 1.f / (1.f + expf(-val));               \
            else if (epi == 2) { float t_ = fmaxf(val, 0.f); *yp = t_ * t_; }\
            else if (epi == 3) *yp += val;                                   \
            else               *yp = val;                                    \
        }                                                                    \
    }
    STORE_TILE(acc0, 0)
    STORE_TILE(acc1, 1)
    STORE_TILE(acc2, 2)
    STORE_TILE(acc3, 3)
    #undef STORE_TILE
}

// ---------------------------------------------------------------------------
extern "C" void kernel_launch(void* const* d_in, const int* in_sizes, int n_in,
                              void* d_out, int out_size, void* d_ws, size_t ws_size,
                              hipStream_t stream) {
    const int*   tokens     = (const int*)  d_in[0];
    const float* emb        = (const float*)d_in[1];
    const float* ln0_w      = (const float*)d_in[2];
    const float* ln0_b      = (const float*)d_in[3];
    const float* ln1_w      = (const float*)d_in[4];
    const float* ln1_b      = (const float*)d_in[5];
    const float* ln2_w      = (const float*)d_in[6];
    const float* ln2_b      = (const float*)d_in[7];
    const float* time_decay = (const float*)d_in[8];
    const float* time_first = (const float*)d_in[9];
    const float* tm_k       = (const float*)d_in[10];
    const float* tm_v       = (const float*)d_in[11];
    const float* tm_r       = (const float*)d_in[12];
    const float* att_key_w  = (const float*)d_in[13];
    const float* att_val_w  = (const float*)d_in[14];
    const float* att_rec_w  = (const float*)d_in[15];
    const float* att_out_w  = (const float*)d_in[16];
    const float* cm_k       = (const float*)d_in[17];
    const float* cm_r       = (const float*)d_in[18];
    const float* ffn_key_w  = (const float*)d_in[19];
    const float* ffn_rec_w  = (const float*)d_in[20];
    const float* ffn_val_w  = (const float*)d_in[21];
    const float* lnout_w    = (const float*)d_in[22];
    const float* lnout_b    = (const float*)d_in[23];
    const float* head_w     = (const float*)d_in[24];
    float* out = (float*)d_out;

    // Workspace layout (fp32): 9 * (M*C) + M*FF = 26M floats (104 MB).
    const size_t MC = (size_t)MROWS * CDIM;
    float* ws  = (float*)d_ws;
    float* X   = ws;          // residual stream [M, C]
    float* XN  = X   + MC;    // LN output       [M, C]
    float* MK  = XN  + MC;    // k-mix           [M, C]
    float* MV  = MK  + MC;    // v-mix           [M, C]
    float* MR  = MV  + MC;    // r-mix           [M, C]
    float* KB  = MR  + MC;    // k               [M, A]
    float* VB  = KB  + MC;    // v / kv          [M, A]
    float* RB  = VB  + MC;    // sigmoid(r)/gate [M, A]
    float* SRY = RB  + MC;    // sigmoid(r)*y    [M, A]
    float* KK  = SRY + MC;    // FFN hidden      [M, FF]

    const dim3 blk(256);
    const int  ew_blocks = (int)(MC / 256);                 // elementwise grid

    embed_ln_kernel<<<MROWS, blk, 0, stream>>>(tokens, emb, ln0_w, ln0_b, X);

    for (int l = 0; l < LAYERS; ++l) {
        // ---- time mix ----
        ln_kernel<<<MROWS, blk, 0, stream>>>(X, ln1_w + l * CDIM, ln1_b + l * CDIM, XN);
        mix_kernel<<<ew_blocks, blk, 0, stream>>>(XN, tm_k + l * CDIM, tm_v + l * CDIM,
                                                  tm_r + l * CDIM, MK, MV, MR, 1);
        gemm_wmma_kernel<<<dim3(ADIM / 128, MROWS / 64), blk, 0, stream>>>(
            MK, att_key_w + (size_t)l * ADIM * CDIM, KB, MROWS, ADIM, CDIM, 0);
        gemm_wmma_kernel<<<dim3(ADIM / 128, MROWS / 64), blk, 0, stream>>>(
            MV, att_val_w + (size_t)l * ADIM * CDIM, VB, MROWS, ADIM, CDIM, 0);
        gemm_wmma_kernel<<<dim3(ADIM / 128, MROWS / 64), blk, 0, stream>>>(
            MR, att_rec_w + (size_t)l * ADIM * CDIM, RB, MROWS, ADIM, CDIM, 1);
        wkv_kernel<<<(BDIM * ADIM) / 256, blk, 0, stream>>>(
            time_decay + l * ADIM, time_first + l * ADIM, KB, VB, RB, SRY);
        gemm_wmma_kernel<<<dim3(CDIM / 128, MROWS / 64), blk, 0, stream>>>(
            SRY, att_out_w + (size_t)l * CDIM * ADIM, X, MROWS, CDIM, ADIM, 3);

        // ---- channel mix ----
        ln_kernel<<<MROWS, blk, 0, stream>>>(X, ln2_w + l * CDIM, ln2_b + l * CDIM, XN);
        mix_kernel<<<ew_blocks, blk, 0, stream>>>(XN, cm_k + l * CDIM, cm_k + l * CDIM,
                                                  cm_r + l * CDIM, MK, MV, MR, 0);
        gemm_wmma_kernel<<<dim3(FFDIM / 128, MROWS / 64), blk, 0, stream>>>(
            MK, ffn_key_w + (size_t)l * FFDIM * CDIM, KK, MROWS, FFDIM, CDIM, 2);
        gemm_wmma_kernel<<<dim3(CDIM / 128, MROWS / 64), blk, 0, stream>>>(
            KK, ffn_val_w + (size_t)l * CDIM * FFDIM, VB, MROWS, CDIM, FFDIM, 0);
        gemm_wmma_kernel<<<dim3(CDIM / 128, MROWS / 64), blk, 0, stream>>>(
            MR, ffn_rec_w + (size_t)l * CDIM * CDIM, RB, MROWS, CDIM, CDIM, 1);
        fma_add_kernel<<<ew_blocks, blk, 0, stream>>>(X, RB, VB);
    }

    ln_kernel<<<MROWS, blk, 0, stream>>>(X, lnout_w, lnout_b, XN);
    gemm_wmma_kernel<<<dim3(VDIM / 128, MROWS / 64), blk, 0, stream>>>(
        XN, head_w, out, MROWS, VDIM, CDIM, 0);
}